// DynamicPatching_47734266527859
// MI455X (gfx1250) — compile-verified
//
#include <hip/hip_runtime.h>

typedef float v2f __attribute__((ext_vector_type(2)));
typedef float v8f __attribute__((ext_vector_type(8)));

#define PATCH 16
#define S_LEN 2048
#define NDIL  4
#define TCNT  2033              // S - PATCH + 1
#define NCAND (NDIL * TCNT)     // 8132
#define PADL  120               // (PATCH-1)*max_dilation = 15*8
#define CH    32                // chunk length per thread for scans
#define NCHK  256               // 256 threads * 32 = 8192 >= 8132

__global__ __launch_bounds__(256)
void dynpatch_fused_kernel(const float* __restrict__ seasonal,
                           const float* __restrict__ conv_w,
                           const float* __restrict__ conv_b,
                           float* __restrict__ out)
{
    __shared__ float xpad[PADL + S_LEN];          // zero-padded signal row
    __shared__ float convs[NDIL][S_LEN];          // 4 dilated conv outputs
    __shared__ float wlds[NDIL * PATCH];
    __shared__ float blds[NDIL];
    __shared__ unsigned int  chunkBase[NCHK + 1]; // exclusive scan of valid counts
    __shared__ unsigned char cm[NCAND];           // can_merge flags
    __shared__ unsigned char mrg[NCAND];          // merge flags after greedy scan
    __shared__ unsigned char so0[NCHK], so1[NCHK], sin[NCHK];

    const int tid  = threadIdx.x;
    const int bIdx = blockIdx.x >> 3;   // batch
    const int cIdx = blockIdx.x & 7;    // channel

    // ---- Phase 0: stage weights + signal row into LDS ----
    if (tid < NDIL * PATCH) wlds[tid] = conv_w[tid];
    if (tid < NDIL)         blds[tid] = conv_b[tid];
    for (int i = tid; i < PADL; i += 256) xpad[i] = 0.0f;
    const float* srow = seasonal + (size_t)bIdx * (S_LEN * 8) + cIdx;
    for (int s = tid; s < S_LEN; s += 256) xpad[PADL + s] = srow[(size_t)s * 8];
    __syncthreads();

    // ---- Phase 1: dilated causal convs via V_WMMA_F32_16X16X4_F32 ----
    // Tile = 16 consecutive outputs of one dilation. A[r][k] = x[t0+r-(15-k)d],
    // B[k][n] = w[k] (weight broadcast over all 16 columns), C preloaded with bias.
    {
        const int lane = tid & 31;
        const int wv   = tid >> 5;          // 8 waves
        const int M    = lane & 15;
        const int g    = lane >> 4;         // K-group per ISA A-layout
        for (int tile = wv; tile < NDIL * (S_LEN / 16); tile += 8) {
            const int di = tile >> 7;           // dilation index
            const int t0 = (tile & 127) << 4;   // output tile start
            const int d  = 1 << di;             // dilation 1,2,4,8
            const float bias = blds[di];
            v8f acc;
            #pragma unroll
            for (int j = 0; j < 8; ++j) acc[j] = bias;
            const int base = PADL + t0 + M;
            #pragma unroll
            for (int c = 0; c < 4; ++c) {       // K=16 split into 4 chunks of 4
                const int k0 = 4 * c + 2 * g;   // lane's K within chunk
                v2f a, b;
                a.x = xpad[base - (15 - k0) * d];
                a.y = xpad[base - (14 - k0) * d];
                b.x = wlds[di * PATCH + k0];
                b.y = wlds[di * PATCH + k0 + 1];
                acc = __builtin_amdgcn_wmma_f32_16x16x4_f32(
                        false, a, false, b, (short)0, acc, false, false);
            }
            // D layout: VGPR j -> rows j (lanes 0-15) and j+8 (lanes 16-31); all columns equal
            if (lane == 0) {
                #pragma unroll
                for (int j = 0; j < 8; ++j) convs[di][t0 + j] = acc[j];
            } else if (lane == 16) {
                #pragma unroll
                for (int j = 0; j < 8; ++j) convs[di][t0 + 8 + j] = acc[j];
            }
        }
    }
    __syncthreads();

    // ---- Phase 2: R^2 merge predicate per adjacent-candidate pair ----
    // pair y = window(i) ++ window(i+1); sxx = 2728, threshold sxy^2 >= 0.5*sxx*syy
    for (int i = tid; i < NCAND - 1; i += 256) {
        const int j2 = i + 1;
        const int b1 = i  / TCNT, t1 = i  - b1 * TCNT;
        const int b2 = j2 / TCNT, t2 = j2 - b2 * TCNT;
        float sy = 0.0f, syy = 0.0f, sxy = 0.0f;
        #pragma unroll
        for (int k = 0; k < PATCH; ++k) {
            const float ya = convs[b1][t1 + k];
            const float yb = convs[b2][t2 + k];
            sy  += ya + yb;
            syy += ya * ya + yb * yb;
            sxy += (float)k * ya + (float)(k + 16) * yb;
        }
        const float sxyc = sxy - 15.5f * sy;
        const float syyc = syy - sy * sy * 0.03125f;
        cm[i] = (syyc > 0.0f) && (sxyc * sxyc >= 1364.0f * syyc) ? 1 : 0;
    }
    if (tid == 0) cm[NCAND - 1] = 0;   // padded False
    __syncthreads();

    // ---- Phase 3: greedy merge recurrence merge[i] = cm[i] & ~merge[i-1] ----
    // Chunked state-composition scan (1-bit transfer function per 32-chunk).
    {
        int st0 = 0, st1 = 1;
        const int i0 = tid * CH;
        for (int k = 0; k < CH; ++k) {
            const int i = i0 + k;
            const int c = (i < NCAND) ? (int)cm[i] : 0;
            st0 = c & (st0 ^ 1);
            st1 = c & (st1 ^ 1);
        }
        so0[tid] = (unsigned char)st0;
        so1[tid] = (unsigned char)st1;
    }
    __syncthreads();
    if (tid == 0) {
        int st = 0;                      // merge[-1] = 0
        for (int c = 0; c < NCHK; ++c) {
            sin[c] = (unsigned char)st;
            st = st ? so1[c] : so0[c];
        }
    }
    __syncthreads();
    {
        int st = sin[tid];
        const int i0 = tid * CH;
        for (int k = 0; k < CH; ++k) {
            const int i = i0 + k;
            if (i < NCAND) {
                const int m = (int)cm[i] & (st ^ 1);
                mrg[i] = (unsigned char)m;
                st = m;
            }
        }
    }
    __syncthreads();

    // ---- Phase 4: compaction positions (valid[i] = i==0 || !merge[i-1]) ----
    {
        int cnt = 0;
        const int i0 = tid * CH;
        for (int k = 0; k < CH; ++k) {
            const int i = i0 + k;
            if (i < NCAND) cnt += (i == 0) ? 1 : ((int)mrg[i - 1] ^ 1);
        }
        so0[tid] = (unsigned char)cnt;   // reuse as per-chunk valid count (<=32)
    }
    __syncthreads();
    if (tid == 0) {
        unsigned int acc = 0;
        for (int c = 0; c < NCHK; ++c) { chunkBase[c] = acc; acc += so0[c]; }
        chunkBase[NCHK] = acc;           // total valid count V
    }
    __syncthreads();

    // ---- Phase 5: write packed output (B, N, p, C) + zero tail ----
    const size_t outRow = (size_t)bIdx * ((size_t)NCAND * PATCH * 8) + (size_t)cIdx;
    {
        unsigned int p = chunkBase[tid];
        const int i0 = tid * CH;
        for (int k = 0; k < CH; ++k) {
            const int i = i0 + k;
            if (i >= NCAND) break;
            const int v = (i == 0) ? 1 : ((int)mrg[i - 1] ^ 1);
            if (v) {
                const int m  = mrg[i];
                const int b1 = i / TCNT, t1 = i - b1 * TCNT;
                int j2 = i + 1; if (j2 > NCAND - 1) j2 = NCAND - 1;
                const int b2 = j2 / TCNT, t2 = j2 - b2 * TCNT;
                float* o = out + outRow + (size_t)p * (PATCH * 8);
                #pragma unroll
                for (int kk = 0; kk < PATCH; ++kk) {
                    float val = convs[b1][t1 + kk];
                    if (m) val = 0.5f * (val + convs[b2][t2 + kk]);
                    o[(size_t)kk * 8] = val;
                }
                ++p;
            }
        }
    }
    {
        const unsigned int V = chunkBase[NCHK];
        for (unsigned int n = V + tid; n < NCAND; n += 256) {
            float* o = out + outRow + (size_t)n * (PATCH * 8);
            #pragma unroll
            for (int kk = 0; kk < PATCH; ++kk) o[(size_t)kk * 8] = 0.0f;
        }
    }
}

extern "C" void kernel_launch(void* const* d_in, const int* in_sizes, int n_in,
                              void* d_out, int out_size, void* d_ws, size_t ws_size,
                              hipStream_t stream)
{
    const float* seasonal = (const float*)d_in[0];  // (32, 2048, 8) f32
    const float* conv_w   = (const float*)d_in[1];  // (4, 16) f32
    const float* conv_b   = (const float*)d_in[2];  // (4,) f32
    float* out = (float*)d_out;                     // (32, 8132, 16, 8) f32

    dynpatch_fused_kernel<<<dim3(32 * 8), dim3(256), 0, stream>>>(
        seasonal, conv_w, conv_b, out);
}